// YoloLoss_72430328480098
// MI455X (gfx1250) — compile-verified
//
#include <hip/hip_runtime.h>
#include <stdint.h>

// Problem constants (match reference)
#define BB      512        // batch
#define NC      1000       // classes
#define GG      64         // 8x8 grid cells
#define KK      8          // boxes per image
#define NCH     1005       // 5 + C channels
#define CH_TILE 100        // channels per TDM tile
#define NT      10         // tiles: 10 * 100 = 1000 channels
#define PLANE   65         // padded LDS stride (floats) per channel plane (64 + 1 pad DWORD)

typedef __attribute__((ext_vector_type(4))) unsigned int u32x4;
typedef __attribute__((ext_vector_type(4))) int          i32x4;
typedef __attribute__((ext_vector_type(8))) int          i32x8;

// Issue one TDM tensor_load_to_lds: 2-D tile of CH_TILE channel planes
// (64 fp32 each) from global -> LDS, with 1-DWORD pad after each 64-DWORD
// plane so LDS plane stride is 65 DWORDs (bank-conflict-free box gathers).
__device__ __forceinline__ void tdm_load_tile(const float* gsrc, unsigned lds_byte) {
  unsigned long long ga = (unsigned long long)(uintptr_t)gsrc;
  u32x4 g0 = { 1u,                                   // count=1, user descriptor
               lds_byte,                             // lds_addr (bytes)
               (unsigned)(ga & 0xFFFFFFFFu),         // global_addr[31:0]
               ((unsigned)(ga >> 32) & 0x01FFFFFFu)  // global_addr[56:32]
                 | 0x80000000u };                    // type = 2 ("image")
  i32x8 g1 = { (2 << 16) | (1 << 20) | (5 << 22),    // data_size=4B, pad_enable,
                                                     // pad_interval=64 DW, pad_amount=1 DW
               (64 << 16),                           // tensor_dim0 = 64
               (CH_TILE << 16),                      // tensor_dim1 = CH_TILE
               (64 << 16),                           // tile_dim0 = 64
               CH_TILE,                              // tile_dim1 = CH_TILE
               64,                                   // tensor_dim0_stride = 64 elems
               0, 0 };
  i32x4 gz = { 0, 0, 0, 0 };
#if defined(__clang_major__) && (__clang_major__ >= 23)
  i32x8 gz8 = { 0, 0, 0, 0, 0, 0, 0, 0 };
  __builtin_amdgcn_tensor_load_to_lds(g0, g1, gz, gz, gz8, 0);
#else
  __builtin_amdgcn_tensor_load_to_lds(g0, g1, gz, gz, 0);
#endif
}

__global__ __launch_bounds__(256) void yolo_loss_main(
    const float* __restrict__ outputs, const float* __restrict__ boxes,
    const int* __restrict__ labels, const int* __restrict__ box_xy,
    const float* __restrict__ obj_target, float* __restrict__ part)
{
  __shared__ float tiles[2][CH_TILE * PLANE];   // 2 x 26 KB double buffer
  __shared__ float sobj[64];
  __shared__ float sbb[32];
  __shared__ float sce[KK];

  const int b    = blockIdx.x;                  // one workgroup per image
  const int tid  = threadIdx.x;                 // 256 threads = 8 waves (wave32)
  const int wave = tid >> 5;
  const int lane = tid & 31;

  const float* img = outputs + (size_t)b * NCH * GG;

  // ---- objectness BCE terms: channel-0 plane (64 contiguous floats) ----
  if (tid < 64) {
    float p   = img[tid];
    float t   = obj_target[b * 64 + tid];
    float lp  = fmaxf(__logf(p),    -100.0f);
    float l1p = fmaxf(log1pf(-p),   -100.0f);
    sobj[tid] = -(t * lp + (1.0f - t) * l1p);
  }
  // ---- bbox MSE terms: channels 1..4 at each box cell ----
  if (tid < 32) {
    int k = tid >> 2, j = tid & 3;
    int cx = box_xy[(b * KK + k) * 2 + 0];
    int cy = box_xy[(b * KK + k) * 2 + 1];
    int cell = cy * 8 + cx;
    float pv  = img[(size_t)(1 + j) * GG + cell];
    float tgt = rintf(boxes[(b * KK + k) * 4 + j] * (10.0f / 256.0f)) * 0.1f;
    float d   = pv - tgt;
    sbb[tid]  = d * d;
  }

  // ---- per-wave box info: wave k owns box k ----
  int cx = box_xy[(b * KK + wave) * 2 + 0];
  int cy = box_xy[(b * KK + wave) * 2 + 1];
  int cell = cy * 8 + cx;
  int lab  = labels[b * KK + wave];                         // 1..NC
  float chosen = img[(size_t)(5 + lab - 1) * GG + cell];    // selected class logit

  const float* cls = img + 5 * GG;                          // class block base
  const unsigned lds_base = (unsigned)(uintptr_t)(void*)&tiles[0][0];

  // Prime first tile (TDM ignores EXEC -> issue from wave 0 only).
  if (wave == 0) {
    tdm_load_tile(cls, lds_base);
    __builtin_amdgcn_s_wait_tensorcnt(0);
  }
  __syncthreads();

  // ---- streaming (online) logsumexp over 1000 classes, from LDS ----
  float m = -__builtin_inff();
  float s = 0.0f;
  for (int t = 0; t < NT; ++t) {
    if (wave == 0 && (t + 1) < NT)
      tdm_load_tile(cls + (size_t)(t + 1) * CH_TILE * GG,
                    lds_base + (unsigned)(((t + 1) & 1) * (CH_TILE * PLANE * 4)));
    const float* tp = tiles[t & 1];
    for (int c = lane; c < CH_TILE; c += 32) {
      float x = tp[c * PLANE + cell];          // banks (c+cell)%64: conflict-free
      if (x > m) { s = s * __expf(m - x) + 1.0f; m = x; }
      else       { s += __expf(x - m); }
    }
    __syncthreads();                           // all done reading tiles[t&1]
    if (wave == 0 && (t + 1) < NT) __builtin_amdgcn_s_wait_tensorcnt(0);
    __syncthreads();                           // next buffer published
  }

  // ---- wave32 merge of (max, sum) pairs ----
  for (int off = 16; off; off >>= 1) {
    float m2 = __shfl_xor(m, off, 32);
    float s2 = __shfl_xor(s, off, 32);
    float mn = fmaxf(m, m2);
    s = s * __expf(m - mn) + s2 * __expf(m2 - mn);
    m = mn;
  }
  if (lane == 0) sce[wave] = (m + __logf(s)) - chosen;
  __syncthreads();

  // ---- fixed-order per-image partials (deterministic) ----
  if (tid == 0) {
    float o = 0.0f;  for (int i = 0; i < 64; ++i) o  += sobj[i];
    float bb = 0.0f; for (int i = 0; i < 32; ++i) bb += sbb[i];
    float ce = 0.0f; for (int i = 0; i < KK; ++i) ce += sce[i];
    part[b * 3 + 0] = o;            // sum of 64 BCE terms
    part[b * 3 + 1] = bb * 0.25f;   // sum over boxes of mean over 4 comps
    part[b * 3 + 2] = ce;           // sum of 8 CE terms
  }
}

__global__ __launch_bounds__(32) void yolo_finalize(const float* __restrict__ part,
                                                    float* __restrict__ out)
{
  const int lane = threadIdx.x;
  float o = 0.0f, bb = 0.0f, ce = 0.0f;
  for (int b = lane; b < BB; b += 32) {
    o  += part[b * 3 + 0];
    bb += part[b * 3 + 1];
    ce += part[b * 3 + 2];
  }
  for (int off = 16; off; off >>= 1) {
    o  += __shfl_xor(o,  off, 32);
    bb += __shfl_xor(bb, off, 32);
    ce += __shfl_xor(ce, off, 32);
  }
  if (lane == 0) {
    float cel_obj   = o  / (float)(BB * GG);
    float bb_loss   = bb / (float)BB;
    float cel_class = ce / (float)BB;
    out[0] = 1.0f * cel_obj + 5.0f * bb_loss + 1.0f * cel_class;
    out[1] = cel_obj;
    out[2] = bb_loss;
    out[3] = cel_class;
  }
}

extern "C" void kernel_launch(void* const* d_in, const int* in_sizes, int n_in,
                              void* d_out, int out_size, void* d_ws, size_t ws_size,
                              hipStream_t stream) {
  const float* outputs    = (const float*)d_in[0];   // (512,1005,8,8) f32
  const float* boxes      = (const float*)d_in[1];   // (512,8,4) f32
  const int*   labels     = (const int*)d_in[2];     // (512,8) i32
  const int*   box_xy     = (const int*)d_in[3];     // (512,8,2) i32
  const float* obj_target = (const float*)d_in[4];   // (512,64) f32
  float* part = (float*)d_ws;                        // 512*3 floats, fully overwritten
  float* out  = (float*)d_out;                       // 4 floats

  yolo_loss_main<<<BB, 256, 0, stream>>>(outputs, boxes, labels, box_xy, obj_target, part);
  yolo_finalize<<<1, 32, 0, stream>>>(part, out);
  (void)in_sizes; (void)n_in; (void)out_size; (void)ws_size;
}